// PotentialAtPoint_34239479283825
// MI455X (gfx1250) — compile-verified
//
#include <hip/hip_runtime.h>
#include <hip/hip_bf16.h>
#include <math.h>

#define NMAX 20

typedef __attribute__((ext_vector_type(2))) float v2f;
typedef __attribute__((ext_vector_type(8))) float v8f;

// ---------------------------------------------------------------------------
// Phase 1: s[i] = sum_b (r0/r)^i, f64 accumulation (B=1e6, i=0..19).
// Wave32 shuffle reduce -> LDS f64 atomics -> global f64 atomics.
// ---------------------------------------------------------------------------
__global__ void reduce_powers_kernel(const float* __restrict__ r,
                                     const float* __restrict__ r0,
                                     double* __restrict__ s, int B) {
    __shared__ double sdata[NMAX];
    if (threadIdx.x < NMAX) sdata[threadIdx.x] = 0.0;
    __syncthreads();

    double acc[NMAX];
#pragma unroll
    for (int i = 0; i < NMAX; ++i) acc[i] = 0.0;

    const int stride = gridDim.x * blockDim.x;
    for (int b = blockIdx.x * blockDim.x + threadIdx.x; b < B; b += stride) {
        const double x = (double)r0[b] / (double)r[b];
        double p = 1.0;
#pragma unroll
        for (int i = 0; i < NMAX; ++i) { acc[i] += p; p *= x; }
    }

#pragma unroll
    for (int i = 0; i < NMAX; ++i) {
        double v = acc[i];
        // wave32 butterfly reduce
        for (int off = 16; off > 0; off >>= 1) v += __shfl_xor(v, off, 32);
        if ((threadIdx.x & 31) == 0) atomicAdd(&sdata[i], v);   // ds_add_f64
    }
    __syncthreads();
    if (threadIdx.x < NMAX) atomicAdd(&s[threadIdx.x], sdata[threadIdx.x]); // global_atomic_add_f64
}

// ---------------------------------------------------------------------------
// Phase 2: pref[k] = (-1)^(k+1) * sqrt(2k-1) * sum_i C[k,i]*s[i]
// C[k,i] = (-1)^(i-1) * exp(lgamma(k+i-1) - lgamma(k-i+1) - 2*lgamma(i)), i<=k
// Single wave; 630 lgammas total — negligible.
// ---------------------------------------------------------------------------
__global__ void pref_kernel(const double* __restrict__ s, float* __restrict__ prefF) {
    const int k0 = threadIdx.x;
    if (k0 >= NMAX) return;
    const int k = k0 + 1;
    double hg = 0.0;
    for (int i = 1; i <= k; ++i) {
        const double lgv = lgamma((double)(k + i - 1))
                         - lgamma((double)(k - i + 1))
                         - 2.0 * lgamma((double)i);
        const double cki = ((i & 1) ? 1.0 : -1.0) * exp(lgv);
        hg += cki * s[i - 1];
    }
    const double sign_k = (k & 1) ? 1.0 : -1.0;
    prefF[k0] = (float)(sign_k * sqrt(2.0 * (double)k - 1.0) * hg);
}

// ---------------------------------------------------------------------------
// Phase 3: out[b] = (sqrt(r0_b)/r_b) * dot(pref, coeffs[b, 0:20])
// via V_WMMA_F32_16X16X4_F32: 16 rows/wave/tile, K=20 -> 5 chained WMMAs.
// A-layout (ISA 7.12.2, 32-bit A 16x4): lanes 0-15 -> K={kc,kc+1},
// lanes 16-31 -> K={kc+2,kc+3}; row scale folded into A so D[:,0] == out.
// ---------------------------------------------------------------------------
__global__ void gemv_wmma_kernel(const float* __restrict__ r,
                                 const float* __restrict__ r0,
                                 const float* __restrict__ coeffs,
                                 const float* __restrict__ pref,
                                 float* __restrict__ out, int B) {
    const int lane         = threadIdx.x & 31;
    const int wave_in_blk  = threadIdx.x >> 5;
    const int waves_per_bk = blockDim.x >> 5;
    const int wave_id      = blockIdx.x * waves_per_bk + wave_in_blk;
    const int n_waves      = gridDim.x * waves_per_bk;

    // pref is 80 B, L2-hot; replicate into registers once per wave.
    float pv[NMAX];
#pragma unroll
    for (int i = 0; i < NMAX; ++i) pv[i] = pref[i];

    const int ntiles = (B + 15) >> 4;
    for (int t = wave_id; t < ntiles; t += n_waves) {
        const int base = t << 4;
        if (base + 16 <= B) {                       // uniform per wave: EXEC all-1s
            const int row   = base + (lane & 15);
            const int khalf = (lane >> 4) << 1;     // 0 (lanes 0-15) or 2 (16-31)

            const double rv  = (double)r[row];
            const double r0v = (double)r0[row];
            const float scale = (float)(sqrt(r0v) / rv);

            v8f c = {};
#pragma unroll
            for (int ch = 0; ch < 5; ++ch) {
                const int kc = 4 * ch;
                const float2 av = *(const float2*)(coeffs + (size_t)row * NMAX + kc + khalf);
                v2f a; a.x = av.x * scale; a.y = av.y * scale;
                // B row k is constant pref[k] across N: every D column = the dot.
                v2f b;
                b.x = (lane < 16) ? pv[kc + 0] : pv[kc + 2];
                b.y = (lane < 16) ? pv[kc + 1] : pv[kc + 3];
                c = __builtin_amdgcn_wmma_f32_16x16x4_f32(
                        false, a, false, b, (short)0, c, false, false);
            }
            // D layout: VGPR j, lanes 0-15 -> M=j (N=lane); lanes 16-31 -> M=j+8.
            // Column N=0 lives in lane 0 (M=0..7) and lane 16 (M=8..15).
            if (lane == 0) {
#pragma unroll
                for (int j = 0; j < 8; ++j) out[base + j] = c[j];
            } else if (lane == 16) {
#pragma unroll
                for (int j = 0; j < 8; ++j) out[base + 8 + j] = c[j];
            }
        } else {                                    // generic tail (unused for B=1e6)
            const int row = base + lane;
            if (row < B) {
                const double rv  = (double)r[row];
                const double r0v = (double)r0[row];
                const float scale = (float)(sqrt(r0v) / rv);
                float acc = 0.0f;
#pragma unroll
                for (int i = 0; i < NMAX; ++i)
                    acc += pv[i] * coeffs[(size_t)row * NMAX + i];
                out[row] = acc * scale;
            }
        }
    }
}

// ---------------------------------------------------------------------------
extern "C" void kernel_launch(void* const* d_in, const int* in_sizes, int n_in,
                              void* d_out, int out_size, void* d_ws, size_t ws_size,
                              hipStream_t stream) {
    const float* r      = (const float*)d_in[0];
    const float* r0     = (const float*)d_in[1];
    const float* coeffs = (const float*)d_in[2];
    float*       out    = (float*)d_out;
    const int B = in_sizes[0];

    double* s     = (double*)d_ws;                              // 20 x f64 accumulators
    float*  prefF = (float*)((char*)d_ws + NMAX * sizeof(double)); // 20 x f32

    hipMemsetAsync(d_ws, 0, NMAX * sizeof(double), stream);

    reduce_powers_kernel<<<512, 256, 0, stream>>>(r, r0, s, B);
    pref_kernel<<<1, 32, 0, stream>>>(s, prefF);

    const int ntiles = (B + 15) >> 4;
    int blocks = (ntiles + 7) / 8;          // 8 waves per 256-thread block
    if (blocks > 2048) blocks = 2048;       // grid-stride covers the rest
    if (blocks < 1) blocks = 1;
    gemv_wmma_kernel<<<blocks, 256, 0, stream>>>(r, r0, coeffs, prefF, out, B);
}